// DglAPPNPNet_33569464386150
// MI455X (gfx1250) — compile-verified
//
#include <hip/hip_runtime.h>
#include <hip/hip_bf16.h>

typedef float v2f __attribute__((ext_vector_type(2)));
typedef float v8f __attribute__((ext_vector_type(8)));

#define D_FEAT 64

// ---------------------------------------------------------------------------
// L2-normalize: one wave32 per node. lane handles cols {lane, lane+32}.
// hn = h / max(||h||, eps);  nrm = max(||h||, eps)  (so h == hn * nrm exactly)
// ---------------------------------------------------------------------------
__global__ void agnn_normalize(const float* __restrict__ h,
                               float* __restrict__ hn,
                               float* __restrict__ nrm, int n) {
  int node = (int)((blockIdx.x * blockDim.x + threadIdx.x) >> 5);
  int lane = threadIdx.x & 31;
  if (node >= n) return;
  const float* row = h + (size_t)node * D_FEAT;
  float x0 = row[lane];
  float x1 = row[lane + 32];
  float ss = x0 * x0 + x1 * x1;
  #pragma unroll
  for (int off = 16; off; off >>= 1) ss += __shfl_xor(ss, off, 32);
  float m = fmaxf(sqrtf(ss), 1e-12f);
  float s = 1.0f / m;
  float* orow = hn + (size_t)node * D_FEAT;
  orow[lane]      = x0 * s;
  orow[lane + 32] = x1 * s;
  if (lane == 0) nrm[node] = m;
}

// Ordered-uint encoding of float for integer atomicMax (monotone total order)
__device__ __forceinline__ unsigned f32_enc(float f) {
  unsigned u = __float_as_uint(f);
  return (u & 0x80000000u) ? ~u : (u | 0x80000000u);
}
__device__ __forceinline__ float f32_dec(unsigned u) {
  return __uint_as_float((u & 0x80000000u) ? (u & 0x7FFFFFFFu) : ~u);
}

// ---------------------------------------------------------------------------
// Per-edge cosine * beta; segment-max into menc (encoded). One wave per edge.
// ---------------------------------------------------------------------------
__global__ void agnn_edge_dot(const float* __restrict__ hn,
                              const int* __restrict__ src,
                              const int* __restrict__ dst,
                              const float* __restrict__ betas, int layer,
                              float* __restrict__ ew,
                              unsigned* __restrict__ menc, int nE) {
  int e = (int)((blockIdx.x * blockDim.x + threadIdx.x) >> 5);
  int lane = threadIdx.x & 31;
  if (e >= nE) return;
  int s = src[e], d = dst[e];
  const float* ps = hn + (size_t)s * D_FEAT;
  const float* pd = hn + (size_t)d * D_FEAT;
  float x = ps[lane] * pd[lane] + ps[lane + 32] * pd[lane + 32];
  #pragma unroll
  for (int off = 16; off; off >>= 1) x += __shfl_xor(x, off, 32);
  if (lane == 0) {
    float ev = betas[layer] * x;
    ew[e] = ev;
    atomicMax(menc + d, f32_enc(ev));
  }
}

// ---------------------------------------------------------------------------
// a = exp(e - m[dst]); z[dst] += a.  One thread per edge.
// ---------------------------------------------------------------------------
__global__ void agnn_edge_exp(float* __restrict__ ew,
                              const int* __restrict__ dst,
                              const unsigned* __restrict__ menc,
                              float* __restrict__ z, int nE) {
  int i = blockIdx.x * blockDim.x + threadIdx.x;
  if (i >= nE) return;
  int d = dst[i];
  float a = expf(ew[i] - f32_dec(menc[d]));
  ew[i] = a;
  unsafeAtomicAdd(z + d, a);
}

// ---------------------------------------------------------------------------
// hnext[dst] += (a/z[dst]) * nrm[src] * hn[src].  One wave per edge.
// (equals alpha * h[src] since h == hn * nrm exactly)
// ---------------------------------------------------------------------------
__global__ void agnn_edge_aggr(const float* __restrict__ hn,
                               const float* __restrict__ nrm,
                               const int* __restrict__ src,
                               const int* __restrict__ dst,
                               const float* __restrict__ ew,
                               const float* __restrict__ z,
                               float* __restrict__ hnext, int nE) {
  int e = (int)((blockIdx.x * blockDim.x + threadIdx.x) >> 5);
  int lane = threadIdx.x & 31;
  if (e >= nE) return;
  int s = src[e], d = dst[e];
  float w = (ew[e] / z[d]) * nrm[s];
  const float* ps = hn + (size_t)s * D_FEAT;
  float* pd = hnext + (size_t)d * D_FEAT;
  unsafeAtomicAdd(pd + lane,      w * ps[lane]);
  unsafeAtomicAdd(pd + lane + 32, w * ps[lane + 32]);
}

// ---------------------------------------------------------------------------
// Final projection out[N,C] = h[N,64] @ W[C,64]^T using V_WMMA_F32_16X16X4_F32.
// One wave per 16x16 output tile; K=64 consumed in 16 steps of 4.
// A frag (16x4 f32, 2 VGPRs): lanes 0-15 -> M=lid, K={k0,k0+1};
//                             lanes 16-31 -> M=lid, K={k0+2,k0+3}.
// B frag (4x16): B[k][n] = W[n][k], mirrored lane/K layout.
// C/D (16x16 f32, 8 VGPRs): VGPR r -> row (half*8 + r), col = lid.
// ---------------------------------------------------------------------------
__global__ void agnn_gemm_wmma(const float* __restrict__ h,
                               const float* __restrict__ W,
                               float* __restrict__ out,
                               int nRows, int nClasses) {
  int wave = (int)((blockIdx.x * blockDim.x + threadIdx.x) >> 5);
  int lane = threadIdx.x & 31;
  int nMT = (nRows + 15) >> 4;
  int nCT = (nClasses + 15) >> 4;
  if (wave >= nMT * nCT) return;            // wave-uniform exit
  int mt = wave / nCT, ct = wave % nCT;
  int m0 = mt << 4, c0 = ct << 4;
  int half = lane >> 4;                     // 0: lanes 0-15, 1: lanes 16-31
  int lid = lane & 15;

  int arow = m0 + lid; if (arow >= nRows) arow = nRows - 1;       // select, no branch
  int bcol = c0 + lid; if (bcol >= nClasses) bcol = nClasses - 1; // select, no branch
  const float* ap = h + (size_t)arow * D_FEAT + half * 2;
  const float* bp = W + (size_t)bcol * D_FEAT + half * 2;

  v8f acc = {0.f, 0.f, 0.f, 0.f, 0.f, 0.f, 0.f, 0.f};
  #pragma unroll
  for (int k0 = 0; k0 < D_FEAT; k0 += 4) {
    v2f a, b;
    a.x = ap[k0]; a.y = ap[k0 + 1];
    b.x = bp[k0]; b.y = bp[k0 + 1];
    acc = __builtin_amdgcn_wmma_f32_16x16x4_f32(false, a, false, b,
                                                (short)0, acc, false, false);
  }

  int orow0 = m0 + half * 8;
  int ocol = c0 + lid;
  if (ocol < nClasses) {
    #pragma unroll
    for (int r = 0; r < 8; ++r) {
      int orow = orow0 + r;
      if (orow < nRows) out[(size_t)orow * nClasses + ocol] = acc[r];
    }
  }
}

// ---------------------------------------------------------------------------
// Host launch
// ---------------------------------------------------------------------------
extern "C" void kernel_launch(void* const* d_in, const int* in_sizes, int n_in,
                              void* d_out, int out_size, void* d_ws, size_t ws_size,
                              hipStream_t stream) {
  const float* features = (const float*)d_in[0];  // [N,64]
  const float* betas    = (const float*)d_in[1];  // [L]
  const float* W        = (const float*)d_in[2];  // [C,64]
  const int*   src      = (const int*)d_in[3];    // [E]
  const int*   dst      = (const int*)d_in[4];    // [E]
  float*       out      = (float*)d_out;          // [N,C]

  const int N = in_sizes[0] / D_FEAT;
  const int L = in_sizes[1];
  const int C = in_sizes[2] / D_FEAT;
  const int E = in_sizes[3];

  // Workspace layout (floats): bufA[N*64] | hn[N*64] | nrm[N] | ew[E] | z[N] | menc[N]
  float* bufA = (float*)d_ws;
  float* hn   = bufA + (size_t)N * D_FEAT;
  float* nrm  = hn + (size_t)N * D_FEAT;
  float* ew   = nrm + N;
  float* z    = ew + E;
  unsigned* menc = (unsigned*)(z + N);

  dim3 blk(256);
  int nodeBlocks = (N + 7) / 8;        // 8 waves per block, wave per node
  int edgeWBlocks = (E + 7) / 8;       // wave per edge
  int edgeTBlocks = (E + 255) / 256;   // thread per edge

  for (int layer = 0; layer < L; ++layer) {
    const float* h_in = (layer == 0) ? features : bufA;
    agnn_normalize<<<nodeBlocks, blk, 0, stream>>>(h_in, hn, nrm, N);
    // h_in (== bufA for layers >= 1) is dead after normalize: safe to reset.
    hipMemsetAsync(bufA, 0, (size_t)N * D_FEAT * sizeof(float), stream);
    hipMemsetAsync(z, 0, (size_t)N * sizeof(float), stream);
    hipMemsetAsync(menc, 0, (size_t)N * sizeof(unsigned), stream);  // encoded -inf
    agnn_edge_dot<<<edgeWBlocks, blk, 0, stream>>>(hn, src, dst, betas, layer, ew, menc, E);
    agnn_edge_exp<<<edgeTBlocks, blk, 0, stream>>>(ew, dst, menc, z, E);
    agnn_edge_aggr<<<edgeWBlocks, blk, 0, stream>>>(hn, nrm, src, dst, ew, z, bufA, E);
  }

  int nMT = (N + 15) >> 4, nCT = (C + 15) >> 4;
  int waves = nMT * nCT;
  agnn_gemm_wmma<<<(waves + 7) / 8, blk, 0, stream>>>(bufA, W, out, N, C);
}